// MiniBeamformerV2_76862734729905
// MI455X (gfx1250) — compile-verified
//
#include <hip/hip_runtime.h>
#include <math.h>

typedef __attribute__((ext_vector_type(2))) float v2f;
typedef __attribute__((ext_vector_type(8))) float v8f;

#define EPSF 1e-4f
#define B_ 2
#define CH_ 8
#define CHCOV_ 64
#define FREQ_ 256
#define T_ 500
#define CHHID_ 24
#define FT_ (FREQ_ * T_) /* 128000 */

static __device__ __forceinline__ v8f wmma4(v2f a, v2f b, v8f c) {
  // D = A(16x4) * B(4x16) + C, fp32. f32 WMMA: NEG[0:1] must be 0, C-mod 0.
  return __builtin_amdgcn_wmma_f32_16x16x4_f32(false, a, false, b, (short)0, c,
                                               false, false);
}

// ---------------------------------------------------------------------------
// Stage 1: per-frequency complex matvec 24x64 over cov via WMMA.
// Each wave computes 2 t-tiles (32 columns) reusing each A (weight) fragment.
// Row guard handled by 0/1 mask multiply (no exec divergence in K-loop).
// ---------------------------------------------------------------------------
__global__ void k_pconv1(const float* __restrict__ cov,
                         const float* __restrict__ pwr,
                         const float* __restrict__ pwi,
                         const float* __restrict__ pbr,
                         const float* __restrict__ pbi,
                         float* __restrict__ c_r, float* __restrict__ c_i) {
  const int bf = blockIdx.x;
  const int b = bf >> 8, f = bf & 255;
  const int wave = threadIdx.x >> 5, lane = threadIdx.x & 31;
  const int half = lane >> 4, ln = lane & 15;
  const int slot = blockIdx.y * 4 + wave;  // 16 slots, 2 tiles each
  const int t0 = slot * 32;

  int tcol[2], tc[2];
#pragma unroll
  for (int n = 0; n < 2; ++n) {
    tcol[n] = t0 + n * 16 + ln;
    tc[n] = tcol[n] < T_ ? tcol[n] : T_ - 1;
  }

  const float* xr0 = cov + (size_t)(b * 2 + 0) * CHCOV_ * FT_ + f * T_;
  const float* xi0 = cov + (size_t)(b * 2 + 1) * CHCOV_ * FT_ + f * T_;
  const float* w0r = pwr + (size_t)(f * CHHID_ + ln) * CHCOV_;
  const float* w0i = pwi + (size_t)(f * CHHID_ + ln) * CHCOV_;
  const int row1c = 16 + (ln < 8 ? ln : 7);           // clamp to valid row
  const float m1 = (ln < 8) ? 1.f : 0.f;              // rows 16..23 valid
  const float* w1r = pwr + (size_t)(f * CHHID_ + row1c) * CHCOV_;
  const float* w1i = pwi + (size_t)(f * CHHID_ + row1c) * CHCOV_;

  v8f aR0[2] = {}, aI0[2] = {}, aR1[2] = {}, aI1[2] = {};
#pragma unroll 4
  for (int ks = 0; ks < 16; ++ks) {
    const int k0 = ks * 4 + half * 2;
    v2f ar0, aw0, ar1, aw1;
    ar0.x = w0r[k0];
    ar0.y = w0r[k0 + 1];
    aw0.x = w0i[k0];
    aw0.y = w0i[k0 + 1];
    ar1.x = w1r[k0] * m1;
    ar1.y = w1r[k0 + 1] * m1;
    aw1.x = w1i[k0] * m1;
    aw1.y = w1i[k0 + 1] * m1;
    const v2f an0 = -aw0, an1 = -aw1;
#pragma unroll
    for (int n = 0; n < 2; ++n) {
      v2f bxr, bxi;
      bxr.x = xr0[(size_t)k0 * FT_ + tc[n]];
      bxr.y = xr0[(size_t)(k0 + 1) * FT_ + tc[n]];
      bxi.x = xi0[(size_t)k0 * FT_ + tc[n]];
      bxi.y = xi0[(size_t)(k0 + 1) * FT_ + tc[n]];
      // Cr = Wr*Xr - Wi*Xi ; Ci = Wr*Xi + Wi*Xr
      aR0[n] = wmma4(ar0, bxr, aR0[n]);
      aR0[n] = wmma4(an0, bxi, aR0[n]);
      aI0[n] = wmma4(ar0, bxi, aI0[n]);
      aI0[n] = wmma4(aw0, bxr, aI0[n]);
      aR1[n] = wmma4(ar1, bxr, aR1[n]);
      aR1[n] = wmma4(an1, bxi, aR1[n]);
      aI1[n] = wmma4(ar1, bxi, aI1[n]);
      aI1[n] = wmma4(aw1, bxr, aI1[n]);
    }
  }
#pragma unroll
  for (int n = 0; n < 2; ++n) {
    if (tcol[n] < T_) {
#pragma unroll
      for (int j = 0; j < 8; ++j) {
        const int r0 = j + (half ? 8 : 0);
        const size_t o0 = (size_t)(b * CHHID_ + r0) * FT_ + f * T_ + tcol[n];
        c_r[o0] = aR0[n][j] + pbr[f * CHHID_ + r0];
        c_i[o0] = aI0[n][j] + pbi[f * CHHID_ + r0];
        const int r1s = 16 + r0;
        if (r1s < CHHID_) {
          const size_t o1 = (size_t)(b * CHHID_ + r1s) * FT_ + f * T_ + tcol[n];
          c_r[o1] = aR1[n][j] + pbr[f * CHHID_ + r1s];
          c_i[o1] = aI1[n][j] + pbi[f * CHHID_ + r1s];
        }
      }
    }
  }
}

// ---------------------------------------------------------------------------
// Stages 2-3 fused: 5-tap causal complex time conv (depthwise) -> tReLU ->
// |.| -> c2 (64x24) -> w-normalized beam weights -> 8x8 complex apply ->
// magnitude modulation.  One thread per (b,f,t).
// ---------------------------------------------------------------------------
__global__ void k_mid(const float* __restrict__ c_r, const float* __restrict__ c_i,
                      const float* __restrict__ tensor,
                      const float* __restrict__ c1b_wr,
                      const float* __restrict__ c1b_wi,
                      const float* __restrict__ c1b_br,
                      const float* __restrict__ c1b_bi,
                      const float* __restrict__ trelu_t,
                      const float* __restrict__ trelu_b,
                      const float* __restrict__ c2_w,
                      const float* __restrict__ c2_b,
                      const float* __restrict__ w,
                      const float* __restrict__ mod_b,
                      float* __restrict__ tf_r, float* __restrict__ tf_i) {
  const int bf = blockIdx.x;
  const int b = bf >> 8, f = bf & 255;
  const int tid = threadIdx.x;
  const int t = blockIdx.y * blockDim.x + tid;

  __shared__ float s_c2w[64 * 24];
  __shared__ float s_c2b[64];
  __shared__ float s_wnr[64], s_wni[64], s_wna[64];
  __shared__ float s_tc[24][6];
  __shared__ float s_cw[24][10];
  __shared__ float s_cb[24][2];
  __shared__ float s_mb[8];

  for (int i = tid; i < 64 * 24; i += blockDim.x) s_c2w[i] = c2_w[i];
  if (tid < 64) {
    s_c2b[tid] = c2_b[tid];
    const float vr = w[tid * FREQ_ + f];
    const float vi = w[64 * FREQ_ + tid * FREQ_ + f];
    const float d = sqrtf(vr * vr + vi * vi + EPSF * EPSF);
    s_wnr[tid] = vr / d;
    s_wni[tid] = vi / d;
    s_wna[tid] = (fabsf(vr) + fabsf(vi)) / d;
  }
  if (tid < 24) {
    const int h = tid;
    s_tc[h][0] = trelu_t[(0 * 24 + h) * FREQ_ + f];
    s_tc[h][1] = trelu_t[(1 * 24 + h) * FREQ_ + f];
    s_tc[h][2] = trelu_b[(0 * 24 + h) * FREQ_ + f];
    s_tc[h][3] = trelu_t[(2 * 24 + h) * FREQ_ + f];
    s_tc[h][4] = trelu_t[(3 * 24 + h) * FREQ_ + f];
    s_tc[h][5] = trelu_b[(1 * 24 + h) * FREQ_ + f];
#pragma unroll
    for (int k = 0; k < 5; ++k) {
      s_cw[h][k] = c1b_wr[h * 5 + k];
      s_cw[h][5 + k] = c1b_wi[h * 5 + k];
    }
    s_cb[h][0] = c1b_br[h];
    s_cb[h][1] = c1b_bi[h];
  }
  if (tid < 8) s_mb[tid] = mod_b[tid * FREQ_ + f];
  __syncthreads();
  if (t >= T_) return;

  float tw2[64];
#pragma unroll
  for (int o = 0; o < 64; ++o) tw2[o] = s_c2b[o];

  for (int h = 0; h < 24; ++h) {
    const float* pr = c_r + (size_t)(b * CHHID_ + h) * FT_ + f * T_;
    const float* pi = c_i + (size_t)(b * CHHID_ + h) * FT_ + f * T_;
    float cr = s_cb[h][0], ci = s_cb[h][1];
#pragma unroll
    for (int k = 0; k < 5; ++k) {
      const int tt = t - 4 + k;
      if (tt >= 0) {
        const float ar = pr[tt], ai = pi[tt];
        const float wr_ = s_cw[h][k], wi_ = s_cw[h][5 + k];
        cr += ar * wr_ - ai * wi_;
        ci += ar * wi_ + ai * wr_;
      }
    }
    float nr = s_tc[h][0] * cr + s_tc[h][1] * ci + s_tc[h][2];
    nr = nr > 0.f ? nr : 0.01f * nr;  // jax leaky_relu default slope
    float ni = s_tc[h][3] * cr + s_tc[h][4] * ci + s_tc[h][5];
    ni = ni > 0.f ? ni : 0.01f * ni;
    const float twh = sqrtf(nr * nr + ni * ni + EPSF * EPSF);
#pragma unroll
    for (int o = 0; o < 64; ++o) tw2[o] += s_c2w[o * 24 + h] * twh;
  }

  float ssum[8];
#pragma unroll
  for (int d = 0; d < 8; ++d) ssum[d] = 0.f;
#pragma unroll
  for (int cc = 0; cc < 64; ++cc) ssum[cc & 7] += fabsf(tw2[cc]) * s_wna[cc];
  float inv[8];
#pragma unroll
  for (int d = 0; d < 8; ++d) inv[d] = 1.f / (ssum[d] + EPSF);

  float xr[8], xi[8];
#pragma unroll
  for (int c = 0; c < 8; ++c) {
    xr[c] = tensor[(size_t)((b * 2 + 0) * CH_ + c) * FT_ + f * T_ + t];
    xi[c] = tensor[(size_t)((b * 2 + 1) * CH_ + c) * FT_ + f * T_ + t];
  }
  float fr[8], fi[8];
#pragma unroll
  for (int d = 0; d < 8; ++d) { fr[d] = 0.f; fi[d] = 0.f; }
#pragma unroll
  for (int c = 0; c < 8; ++c) {
#pragma unroll
    for (int d = 0; d < 8; ++d) {
      const int cc = c * 8 + d;
      const float tr = tw2[cc] * s_wnr[cc] * inv[d];
      const float ti = tw2[cc] * s_wni[cc] * inv[d];
      fr[d] += xr[c] * tr - xi[c] * ti;
      fi[d] += xi[c] * tr + xr[c] * ti;
    }
  }
#pragma unroll
  for (int d = 0; d < 8; ++d) {
    const float mag = sqrtf(fr[d] * fr[d] + fi[d] * fi[d] + EPSF * EPSF);
    const float sc = fmaxf(mag + s_mb[d], 0.f) / mag;
    const size_t o = (size_t)(b * CH_ + d) * FT_ + f * T_ + t;
    tf_r[o] = fr[d] * sc;
    tf_i[o] = fi[d] * sc;
  }
}

// ---------------------------------------------------------------------------
// Stage 4: dense complex conv over frequency, 16ch(=tf concat tensor)->16ch,
// k=3, zero-padded.  K = 48 -> 12 WMMA K-steps.  Weights are f-independent:
// preload ALL A-fragments into registers once, then sweep 4 t-tiles.
// ---------------------------------------------------------------------------
__global__ void k_c3a(const float* __restrict__ tf_r, const float* __restrict__ tf_i,
                      const float* __restrict__ tensor,
                      const float* __restrict__ cwr, const float* __restrict__ cwi,
                      const float* __restrict__ cbr, const float* __restrict__ cbi,
                      float* __restrict__ y_r, float* __restrict__ y_i) {
  const int bf = blockIdx.x;
  const int b = bf >> 8, f = bf & 255;
  const int wave = threadIdx.x >> 5, lane = threadIdx.x & 31;
  const int half = lane >> 4, ln = lane & 15;
  const int slot = blockIdx.y * 4 + wave;  // 8 slots, 4 tiles each

  // Preload all A fragments (weights independent of f and t).
  v2f Ar[12], Aw[12];
#pragma unroll
  for (int ks = 0; ks < 12; ++ks) {
    const int k0 = ks * 4 + half * 2;
    const int ci0 = k0 / 3, kf0 = k0 - ci0 * 3;
    const int ci1 = (k0 + 1) / 3, kf1 = (k0 + 1) - ci1 * 3;
    Ar[ks].x = cwr[(ln * 16 + ci0) * 3 + kf0];
    Ar[ks].y = cwr[(ln * 16 + ci1) * 3 + kf1];
    Aw[ks].x = cwi[(ln * 16 + ci0) * 3 + kf0];
    Aw[ks].y = cwi[(ln * 16 + ci1) * 3 + kf1];
  }
  const float bre = cbr[ln] /*unused by stores directly*/;
  (void)bre;

#pragma unroll
  for (int nt = 0; nt < 4; ++nt) {
    const int t0 = (slot * 4 + nt) * 16;
    const int tcol = t0 + ln;
    const int tc = tcol < T_ ? tcol : T_ - 1;
    v8f aR = {}, aI = {};
#pragma unroll
    for (int ks = 0; ks < 12; ++ks) {
      float bvr[2], bvi[2];
#pragma unroll
      for (int e = 0; e < 2; ++e) {
        const int k = ks * 4 + half * 2 + e;
        const int ci = k / 3, kf = k - ci * 3;
        const int fs = f + kf - 1;
        const bool ok = (unsigned)fs < (unsigned)FREQ_;
        const int fsc = ok ? fs : f;  // clamped valid address
        float vr, vi;
        if (ci < CH_) {
          const size_t o = (size_t)(b * CH_ + ci) * FT_ + fsc * T_ + tc;
          vr = tf_r[o];
          vi = tf_i[o];
        } else {
          vr = tensor[(size_t)((b * 2 + 0) * CH_ + (ci - CH_)) * FT_ + fsc * T_ + tc];
          vi = tensor[(size_t)((b * 2 + 1) * CH_ + (ci - CH_)) * FT_ + fsc * T_ + tc];
        }
        bvr[e] = ok ? vr : 0.f;
        bvi[e] = ok ? vi : 0.f;
      }
      v2f Bx, By;
      Bx.x = bvr[0]; Bx.y = bvr[1];
      By.x = bvi[0]; By.y = bvi[1];
      aR = wmma4(Ar[ks], Bx, aR);
      aR = wmma4(-Aw[ks], By, aR);
      aI = wmma4(Ar[ks], By, aI);
      aI = wmma4(Aw[ks], Bx, aI);
    }
    if (tcol < T_) {
#pragma unroll
      for (int j = 0; j < 8; ++j) {
        const int o = j + (half ? 8 : 0);
        const size_t oo = (size_t)(b * 16 + o) * FT_ + f * T_ + tcol;
        y_r[oo] = aR[j] + cbr[o];
        y_i[oo] = aI[j] + cbi[o];
      }
    }
  }
}

// ---------------------------------------------------------------------------
// Stage 5: final per-frequency complex GEMM, 8 out x (16ch * 5 causal taps),
// K = 80 -> 20 WMMA K-steps.  Per-f weights preloaded once (80 VGPRs),
// then sweep 4 t-tiles issuing only B loads + WMMAs.
// ---------------------------------------------------------------------------
__global__ void k_pconv3(const float* __restrict__ y_r, const float* __restrict__ y_i,
                         const float* __restrict__ pwr, const float* __restrict__ pwi,
                         const float* __restrict__ pbr, const float* __restrict__ pbi,
                         float* __restrict__ out) {
  const int bf = blockIdx.x;
  const int b = bf >> 8, f = bf & 255;
  const int wave = threadIdx.x >> 5, lane = threadIdx.x & 31;
  const int half = lane >> 4, ln = lane & 15;
  const int slot = blockIdx.y * 4 + wave;  // 8 slots, 4 tiles each

  const float mrow = (ln < CH_) ? 1.f : 0.f;
  const int rowc = ln < CH_ ? ln : CH_ - 1;  // clamped valid row
  const float* wrp = pwr + (size_t)(f * CH_ + rowc) * 16 * 5;
  const float* wip = pwi + (size_t)(f * CH_ + rowc) * 16 * 5;

  v2f Ar[20], Aw[20];
#pragma unroll
  for (int ks = 0; ks < 20; ++ks) {
    const int k0 = ks * 4 + half * 2;
    Ar[ks].x = wrp[k0] * mrow;       // (c,kk) flattens to k = c*5+kk
    Ar[ks].y = wrp[k0 + 1] * mrow;
    Aw[ks].x = wip[k0] * mrow;
    Aw[ks].y = wip[k0 + 1] * mrow;
  }

#pragma unroll
  for (int nt = 0; nt < 4; ++nt) {
    const int t0 = (slot * 4 + nt) * 16;
    const int tcol = t0 + ln;
    const int tc = tcol < T_ ? tcol : T_ - 1;
    v8f aR = {}, aI = {};
#pragma unroll
    for (int ks = 0; ks < 20; ++ks) {
      float bvr[2], bvi[2];
#pragma unroll
      for (int e = 0; e < 2; ++e) {
        const int k = ks * 4 + half * 2 + e;
        const int c = k / 5, kk = k - c * 5;
        const int ts = tc + kk - 4;  // causal window
        const bool ok = ts >= 0;
        const int tsc = ok ? ts : 0;
        const size_t o = (size_t)(b * 16 + c) * FT_ + f * T_ + tsc;
        const float vr = y_r[o];
        const float vi = y_i[o];
        bvr[e] = ok ? vr : 0.f;
        bvi[e] = ok ? vi : 0.f;
      }
      v2f Bx, By;
      Bx.x = bvr[0]; Bx.y = bvr[1];
      By.x = bvi[0]; By.y = bvi[1];
      aR = wmma4(Ar[ks], Bx, aR);
      aR = wmma4(-Aw[ks], By, aR);
      aI = wmma4(Ar[ks], By, aI);
      aI = wmma4(Aw[ks], Bx, aI);
    }
    if (tcol < T_) {
#pragma unroll
      for (int j = 0; j < 8; ++j) {
        const int row = j + (half ? 8 : 0);
        if (row < CH_) {
          out[(size_t)((b * 2 + 0) * CH_ + row) * FT_ + f * T_ + tcol] =
              aR[j] + pbr[f * CH_ + row];
          out[(size_t)((b * 2 + 1) * CH_ + row) * FT_ + f * T_ + tcol] =
              aI[j] + pbi[f * CH_ + row];
        }
      }
    }
  }
}

// ---------------------------------------------------------------------------
extern "C" void kernel_launch(void* const* d_in, const int* in_sizes, int n_in,
                              void* d_out, int out_size, void* d_ws, size_t ws_size,
                              hipStream_t stream) {
  const float* tensor = (const float*)d_in[0];
  const float* cov = (const float*)d_in[1];
  const float* pc1_wr = (const float*)d_in[2];
  const float* pc1_wi = (const float*)d_in[3];
  const float* pc1_br = (const float*)d_in[4];
  const float* pc1_bi = (const float*)d_in[5];
  const float* c1b_wr = (const float*)d_in[6];
  const float* c1b_wi = (const float*)d_in[7];
  const float* c1b_br = (const float*)d_in[8];
  const float* c1b_bi = (const float*)d_in[9];
  const float* trelu_t = (const float*)d_in[10];
  const float* trelu_b = (const float*)d_in[11];
  const float* c2_w = (const float*)d_in[12];
  const float* c2_b = (const float*)d_in[13];
  const float* w = (const float*)d_in[14];
  const float* mod_b = (const float*)d_in[15];
  const float* c3a_wr = (const float*)d_in[16];
  const float* c3a_wi = (const float*)d_in[17];
  const float* c3a_br = (const float*)d_in[18];
  const float* c3a_bi = (const float*)d_in[19];
  const float* pc3_wr = (const float*)d_in[20];
  const float* pc3_wi = (const float*)d_in[21];
  const float* pc3_br = (const float*)d_in[22];
  const float* pc3_bi = (const float*)d_in[23];

  float* ws = (float*)d_ws;
  // layout (floats): c_r[6144000] c_i[6144000] tf_r[2048000] tf_i[2048000]
  float* c_r = ws;
  float* c_i = ws + 6144000;
  float* tf_r = ws + 12288000;
  float* tf_i = ws + 14336000;
  // y (B,16,F,T)=4096000 each reuses the dead c_r / c_i regions
  float* y_r = ws;
  float* y_i = ws + 6144000;
  float* outp = (float*)d_out;

  dim3 blk(128, 1, 1);
  dim3 grd1(B_ * FREQ_, 4, 1);  // 16 wave-slots * 2 t-tiles = 32 tiles of 16
  dim3 grd4(B_ * FREQ_, 2, 1);  // 8 wave-slots * 4 t-tiles = 32 tiles of 16
  dim3 grdM(B_ * FREQ_, 4, 1);  // 4 * 128 threads = 512 t (covers 500)

  k_pconv1<<<grd1, blk, 0, stream>>>(cov, pc1_wr, pc1_wi, pc1_br, pc1_bi, c_r, c_i);
  k_mid<<<grdM, blk, 0, stream>>>(c_r, c_i, tensor, c1b_wr, c1b_wi, c1b_br, c1b_bi,
                                  trelu_t, trelu_b, c2_w, c2_b, w, mod_b, tf_r, tf_i);
  k_c3a<<<grd4, blk, 0, stream>>>(tf_r, tf_i, tensor, c3a_wr, c3a_wi, c3a_br,
                                  c3a_bi, y_r, y_i);
  k_pconv3<<<grd4, blk, 0, stream>>>(y_r, y_i, pc3_wr, pc3_wi, pc3_br, pc3_bi, outp);
}